// WeightedHeteroEncoder_56521769616068
// MI455X (gfx1250) — compile-verified
//
#include <hip/hip_runtime.h>
#include <hip/hip_bf16.h>

// ---------------------------------------------------------------------------
// WeightedHeteroEncoder for MI455X (gfx1250, wave32).
// Dense projections use V_WMMA_F32_16X16X4_F32 (full-precision fp32 WMMA).
// Edge phase: one wave per edge, float4 gathers, shfl head reductions,
// segment softmax via monotone-uint atomicMax + float atomicAdd.
// ---------------------------------------------------------------------------

#define N_RNA 50000
#define N_DIS 2000
#define E_RD  200000
#define E_DR  200000
#define E_RR  200000
#define D_IN  256
#define D_DIS 128
#define HC    256   // H*C
#define HEADS 4
#define CHAN  64
#define NEG_SLOPE 0.2f

typedef float v2f __attribute__((ext_vector_type(2)));
typedef float v8f __attribute__((ext_vector_type(8)));

// -------------------- monotone float<->uint order encoding -----------------
__device__ __forceinline__ unsigned fenc(float f) {
    unsigned u = __float_as_uint(f);
    return (u & 0x80000000u) ? ~u : (u | 0x80000000u);
}
__device__ __forceinline__ float fdec(unsigned e) {
    return (e & 0x80000000u) ? __uint_as_float(e & 0x7FFFFFFFu)
                             : __uint_as_float(~e);
}
#define ENC_NEG_INF 0x007FFFFFu   // fenc(-inf)

// ---------------------------------------------------------------------------
// GEMM + bias:  C[M,N] = A[M,K] @ W[K,N] + bias[N]
// One wave computes one 16x16 tile of C via K/4 chained v_wmma_f32_16x16x4_f32.
// A-fragment per ISA layout: lane l holds M=l&15, K = 2*(l>>4) + {0,1} (v2f).
// B-fragment: lane l holds N=l&15, K = 2*(l>>4) + {0,1}.
// C/D: VGPR r, lane l -> row = r + 8*(l>>4), col = l&15.
// Requires M%16==0, N%16==0, K%4==0 (true for all shapes here).
// ---------------------------------------------------------------------------
__global__ void gemm_bias_wmma_f32(const float* __restrict__ A,
                                   const float* __restrict__ W,
                                   const float* __restrict__ bias,
                                   float* __restrict__ C,
                                   int M, int K, int N) {
    const int wave   = blockIdx.x * (blockDim.x >> 5) + (threadIdx.x >> 5);
    const int tilesN = N >> 4;
    const int tiles  = (M >> 4) * tilesN;
    if (wave >= tiles) return;                 // wave-uniform: EXEC stays all-1s

    const int tm   = (wave / tilesN) << 4;
    const int tn   = (wave % tilesN) << 4;
    const int lane = threadIdx.x & 31;
    const int half = lane >> 4;                // 0 or 1
    const int l15  = lane & 15;
    const int koff = half << 1;                // K sub-offset {0,2}

    const float* Arow = A + (size_t)(tm + l15) * K;     // row m = l15
    const float* Wcol = W + (tn + l15);                 // col n = l15

    v8f acc = {};
    for (int k0 = 0; k0 < K; k0 += 4) {
        v2f a = *(const v2f*)(Arow + k0 + koff);        // contiguous in K
        v2f b;
        b.x = Wcol[(size_t)(k0 + koff)     * N];
        b.y = Wcol[(size_t)(k0 + koff + 1) * N];
        acc = __builtin_amdgcn_wmma_f32_16x16x4_f32(
                  false, a, false, b, (short)0, acc, false, false);
    }

    const float bv = bias[tn + l15];
#pragma unroll
    for (int r = 0; r < 8; ++r) {
        int m = tm + r + (half << 3);
        C[(size_t)m * N + tn + l15] = acc[r] + bv;
    }
}

// ---------------------------------------------------------------------------
// Init output accumulator rows with conv bias (b2 optional, summed).
// ---------------------------------------------------------------------------
__global__ void init_bias_rows(float* __restrict__ out,
                               const float* __restrict__ b1,
                               const float* __restrict__ b2,
                               int rows) {
    int t = blockIdx.x * blockDim.x + threadIdx.x;
    if (t >= rows * HC) return;
    int j = t & (HC - 1);
    float v = b1[j];
    if (b2) v += b2[j];
    out[t] = v;
}

__global__ void init_softmax_state(unsigned* __restrict__ amax,
                                   float* __restrict__ den, int n) {
    int t = blockIdx.x * blockDim.x + threadIdx.x;
    if (t >= n) return;
    amax[t] = ENC_NEG_INF;
    den[t]  = 0.0f;
}

// ---------------------------------------------------------------------------
// Edge pass 1: alpha[e,h] = att_h . leaky_relu(xl[src] + xr[dst] + w*We)
// One wave per edge; lane handles 8 contiguous channels (lanes 0-7 -> head 0,
// 8-15 -> head 1, ...). Per-head reduction over 8 lanes with shfl_xor.
// Also does segment-max via encoded atomicMax.
// ---------------------------------------------------------------------------
#define LRELU_TERM(XL, XR, WE, AT)                                   \
    { float m_ = (XL) + (XR) + wt * (WE);                            \
      m_ = (m_ > 0.0f) ? m_ : NEG_SLOPE * m_;                        \
      p += m_ * (AT); }

__global__ void edge_alpha_kernel(const float* __restrict__ xl,
                                  const float* __restrict__ xr,
                                  const int* __restrict__ src,
                                  const int* __restrict__ dst,
                                  const float* __restrict__ w,
                                  const float* __restrict__ We,   // [HC]
                                  const float* __restrict__ att,  // [H*C] flat
                                  float* __restrict__ alpha,      // [E,H]
                                  unsigned* __restrict__ amax,    // [Nd,H]
                                  int E) {
    int e = blockIdx.x * (blockDim.x >> 5) + (threadIdx.x >> 5);
    if (e >= E) return;
    int lane = threadIdx.x & 31;
    int j0 = lane << 3;                         // 8 channels per lane
    int s = src[e], d = dst[e];
    float wt = w[e];

    const float* xls = xl + (size_t)s * HC + j0;
    const float* xrd = xr + (size_t)d * HC + j0;
    float4 a0 = *(const float4*)(xls);
    float4 a1 = *(const float4*)(xls + 4);
    float4 r0 = *(const float4*)(xrd);
    float4 r1 = *(const float4*)(xrd + 4);
    float4 w0 = *(const float4*)(We + j0);
    float4 w1 = *(const float4*)(We + j0 + 4);
    float4 t0 = *(const float4*)(att + j0);
    float4 t1 = *(const float4*)(att + j0 + 4);

    float p = 0.0f;
    LRELU_TERM(a0.x, r0.x, w0.x, t0.x)
    LRELU_TERM(a0.y, r0.y, w0.y, t0.y)
    LRELU_TERM(a0.z, r0.z, w0.z, t0.z)
    LRELU_TERM(a0.w, r0.w, w0.w, t0.w)
    LRELU_TERM(a1.x, r1.x, w1.x, t1.x)
    LRELU_TERM(a1.y, r1.y, w1.y, t1.y)
    LRELU_TERM(a1.z, r1.z, w1.z, t1.z)
    LRELU_TERM(a1.w, r1.w, w1.w, t1.w)

    // reduce within each group of 8 lanes (one head per group)
    p += __shfl_xor(p, 1, 32);
    p += __shfl_xor(p, 2, 32);
    p += __shfl_xor(p, 4, 32);

    if ((lane & 7) == 0) {
        int h = lane >> 3;
        alpha[(size_t)e * HEADS + h] = p;
        atomicMax(&amax[(size_t)d * HEADS + h], fenc(p));
    }
}

// ---------------------------------------------------------------------------
// Edge pass 2: ex = exp(alpha - amax[dst]); den[dst] += ex. 1 thread per (e,h).
// ---------------------------------------------------------------------------
__global__ void edge_exp_kernel(const float* __restrict__ alpha,
                                const int* __restrict__ dst,
                                const unsigned* __restrict__ amax,
                                float* __restrict__ ex,
                                float* __restrict__ den,
                                int E4) {
    int t = blockIdx.x * blockDim.x + threadIdx.x;
    if (t >= E4) return;
    int e = t >> 2, h = t & 3;
    int d = dst[e];
    float v = expf(alpha[t] - fdec(amax[(size_t)d * HEADS + h]));
    ex[t] = v;
    atomicAdd(&den[(size_t)d * HEADS + h], v);
}

// ---------------------------------------------------------------------------
// Edge pass 3: out[dst] += xl[src] * (ex / (den[dst] + 1e-16)). Wave per edge.
// ---------------------------------------------------------------------------
__global__ void edge_scatter_kernel(const float* __restrict__ xl,
                                    const float* __restrict__ ex,
                                    const float* __restrict__ den,
                                    const int* __restrict__ src,
                                    const int* __restrict__ dst,
                                    float* __restrict__ out,
                                    int E) {
    int e = blockIdx.x * (blockDim.x >> 5) + (threadIdx.x >> 5);
    if (e >= E) return;
    int lane = threadIdx.x & 31;
    int j0 = lane << 3;
    int h = lane >> 3;
    int s = src[e], d = dst[e];
    float a = ex[(size_t)e * HEADS + h] /
              (den[(size_t)d * HEADS + h] + 1e-16f);

    const float* xls = xl + (size_t)s * HC + j0;
    float* outd = out + (size_t)d * HC + j0;
    float4 v0 = *(const float4*)(xls);
    float4 v1 = *(const float4*)(xls + 4);
    atomicAdd(outd + 0, v0.x * a);
    atomicAdd(outd + 1, v0.y * a);
    atomicAdd(outd + 2, v0.z * a);
    atomicAdd(outd + 3, v0.w * a);
    atomicAdd(outd + 4, v1.x * a);
    atomicAdd(outd + 5, v1.y * a);
    atomicAdd(outd + 6, v1.z * a);
    atomicAdd(outd + 7, v1.w * a);
}

// ---------------------------------------------------------------------------
// Host-side orchestration
// ---------------------------------------------------------------------------
static inline void launch_gemm(const float* A, const float* W, const float* b,
                               float* C, int M, int K, int N, hipStream_t s) {
    int tiles  = (M / 16) * (N / 16);
    int wpb    = 4;                                   // 4 waves / block
    int blocks = (tiles + wpb - 1) / wpb;
    gemm_bias_wmma_f32<<<blocks, wpb * 32, 0, s>>>(A, W, b, C, M, K, N);
}

extern "C" void kernel_launch(void* const* d_in, const int* in_sizes, int n_in,
                              void* d_out, int out_size, void* d_ws, size_t ws_size,
                              hipStream_t stream) {
    // ---- inputs, in setup_inputs() order ----
    const float* x_rna    = (const float*)d_in[0];
    const float* x_dis    = (const float*)d_in[1];
    const float* w_edge[3] = { (const float*)d_in[2],   // w_rd
                               (const float*)d_in[3],   // w_dr
                               (const float*)d_in[4] }; // w_rr
    const float* lin_d_W  = (const float*)d_in[5];
    const float* lin_d_b  = (const float*)d_in[6];
    // per-relation params: Wl, bl, Wr, br, We, att, bias  (7 each)
    const float* rel[3][7];
    for (int r = 0; r < 3; ++r)
        for (int k = 0; k < 7; ++k)
            rel[r][k] = (const float*)d_in[7 + r * 7 + k];
    const float* lin_out_W = (const float*)d_in[28];
    const float* lin_out_b = (const float*)d_in[29];
    const int* ei_src[3] = { (const int*)d_in[30], (const int*)d_in[32],
                             (const int*)d_in[34] };
    const int* ei_dst[3] = { (const int*)d_in[31], (const int*)d_in[33],
                             (const int*)d_in[35] };
    const int  E[3]     = { E_RD, E_DR, E_RR };
    const int  n_src[3] = { N_RNA, N_DIS, N_RNA };   // (unused, doc only)
    const int  n_dst[3] = { N_DIS, N_RNA, N_RNA };
    (void)n_src; (void)n_in; (void)in_sizes; (void)out_size; (void)ws_size;

    // ---- workspace carve-up (floats) ----
    float* ws    = (float*)d_ws;
    float* xd    = ws;                               // [N_DIS , HC]
    float* xl    = xd    + (size_t)N_DIS * HC;       // [N_RNA , HC]
    float* xr    = xl    + (size_t)N_RNA * HC;       // [N_RNA , HC]
    float* h_rna = xr    + (size_t)N_RNA * HC;       // [N_RNA , HC]
    float* h_dis = h_rna + (size_t)N_RNA * HC;       // [N_DIS , HC]
    float* alpha = h_dis + (size_t)N_DIS * HC;       // [Emax  , H]
    float* exv   = alpha + (size_t)E_RR  * HEADS;    // [Emax  , H]
    float* den   = exv   + (size_t)E_RR  * HEADS;    // [N_RNA , H]
    unsigned* amax = (unsigned*)(den + (size_t)N_RNA * HEADS); // [N_RNA, H]

    // 1) xd = x_dis @ lin_d_W + lin_d_b        [2000,128]x[128,256]
    launch_gemm(x_dis, lin_d_W, lin_d_b, xd, N_DIS, D_DIS, HC, stream);

    // 2) bias-initialize accumulators (conv biases folded in)
    {
        int t = N_DIS * HC;
        init_bias_rows<<<(t + 255) / 256, 256, 0, stream>>>(
            h_dis, rel[0][6], nullptr, N_DIS);            // rd_bias
        t = N_RNA * HC;
        init_bias_rows<<<(t + 255) / 256, 256, 0, stream>>>(
            h_rna, rel[1][6], rel[2][6], N_RNA);          // dr_bias + rr_bias
    }

    // 3) three GATv2 relations, sequential, reusing xl/xr/softmax buffers
    const float* srcfeat[3] = { x_rna, xd,    x_rna };
    const float* dstfeat[3] = { xd,    x_rna, x_rna };
    const int    Ms[3]      = { N_RNA, N_DIS, N_RNA };
    const int    Md[3]      = { N_DIS, N_RNA, N_RNA };
    float*       outacc[3]  = { h_dis, h_rna, h_rna };

    for (int r = 0; r < 3; ++r) {
        const float* Wl  = rel[r][0]; const float* bl  = rel[r][1];
        const float* Wr  = rel[r][2]; const float* br  = rel[r][3];
        const float* We  = rel[r][4]; const float* att = rel[r][5];

        launch_gemm(srcfeat[r], Wl, bl, xl, Ms[r], D_IN, HC, stream);  // xl
        launch_gemm(dstfeat[r], Wr, br, xr, Md[r], D_IN, HC, stream);  // xr

        int nseg = n_dst[r] * HEADS;
        init_softmax_state<<<(nseg + 255) / 256, 256, 0, stream>>>(amax, den, nseg);

        int eblocks = (E[r] + 7) / 8;                 // 8 edges (waves) / block
        edge_alpha_kernel<<<eblocks, 256, 0, stream>>>(
            xl, xr, ei_src[r], ei_dst[r], w_edge[r], We, att, alpha, amax, E[r]);

        int e4 = E[r] * HEADS;
        edge_exp_kernel<<<(e4 + 255) / 256, 256, 0, stream>>>(
            alpha, ei_dst[r], amax, exv, den, e4);

        edge_scatter_kernel<<<eblocks, 256, 0, stream>>>(
            xl, exv, den, ei_src[r], ei_dst[r], outacc[r], E[r]);
    }

    // 4) output projection into d_out = [h_rna ; h_dis] @ lin_out_W + b
    float* out_f = (float*)d_out;
    launch_gemm(h_rna, lin_out_W, lin_out_b, out_f,                 N_RNA, HC, CHAN, stream);
    launch_gemm(h_dis, lin_out_W, lin_out_b, out_f + (size_t)N_RNA * CHAN, N_DIS, HC, CHAN, stream);
}